// ElementUpdate_78134045049160
// MI455X (gfx1250) — compile-verified
//
#include <hip/hip_runtime.h>

typedef __attribute__((ext_vector_type(2))) float v2f;
typedef __attribute__((ext_vector_type(8))) float v8f;

#define HIDDEN 128

// One block = 256 threads = 8 wave32 waves. Block b owns node tile [16b, 16b+16).
// Wave w owns output columns [16w, 16w+16). Each wave computes the 16x16 output
// tile O = M_tile x W[s]^T with 32 chained V_WMMA_F32_16X16X4_F32 ops (K=128).
// atom_types is sorted, so a 16-row tile usually holds one species; a wave-uniform
// loop handles the rare multi-species tiles, predicating only the stores.
__global__ __launch_bounds__(256)
void element_update_wmma(const float* __restrict__ h_prev,
                         const float* __restrict__ m_curr,
                         const int*   __restrict__ atom_types,
                         const float* __restrict__ weight,
                         float*       __restrict__ out)
{
    const int lane  = threadIdx.x & 31;
    const int wave  = threadIdx.x >> 5;        // 0..7 -> output column tile
    const int row   = lane & 15;               // matrix row handled by this lane
    const int kk    = (lane >> 4) << 1;        // K sub-pair: 0 (lanes 0-15) or 2 (16-31)
    const int hi8   = (lane >> 4) << 3;        // C/D row offset: 0 or 8
    const int node0 = blockIdx.x << 4;

    const int col = (wave << 4) + row;         // output column for B fragment / stores
    const float* Mrow = m_curr + (size_t)(node0 + row) * HIDDEN;

    const int myT = atom_types[node0 + row];   // lanes 0-15 cover the 16 tile rows
    int t = __shfl(myT, 0, 32);                // first (smallest) species in tile

    for (;;) {
        // W[t] row for this lane's output column: B[j,i] = W[i, k+j] (O = M * W^T)
        const float* Wrow = weight + ((size_t)t * HIDDEN + col) * HIDDEN;

        v8f acc = {};
        #pragma unroll
        for (int k = 0; k < HIDDEN; k += 4) {
            v2f a = *(const v2f*)(Mrow + k + kk);   // A: M[row, k+kk .. k+kk+1]
            v2f b = *(const v2f*)(Wrow + k + kk);   // B: W[col, k+kk .. k+kk+1]
            acc = __builtin_amdgcn_wmma_f32_16x16x4_f32(
                /*neg_a=*/false, a, /*neg_b=*/false, b,
                /*c_mod=*/(short)0, acc, /*reuse_a=*/false, /*reuse_b=*/false);
        }

        // Writeback: acc[r] -> O[node0 + r + hi8, col]; only rows of species t.
        #pragma unroll
        for (int r = 0; r < 8; ++r) {
            int rt = __shfl(myT, r + hi8, 32);
            if (rt == t) {
                size_t idx = (size_t)(node0 + r + hi8) * HIDDEN + col;
                out[idx] = h_prev[idx] + acc[r];
            }
        }

        // Next species strictly greater than t present in this tile (wave-uniform).
        int cand = (myT > t) ? myT : 0x7fffffff;
        #pragma unroll
        for (int off = 16; off; off >>= 1)
            cand = min(cand, __shfl_xor(cand, off, 32));
        if (cand == 0x7fffffff) break;
        t = cand;
    }
}

extern "C" void kernel_launch(void* const* d_in, const int* in_sizes, int n_in,
                              void* d_out, int out_size, void* d_ws, size_t ws_size,
                              hipStream_t stream) {
    const float* h_prev     = (const float*)d_in[0];
    const float* m_curr     = (const float*)d_in[1];
    const int*   atom_types = (const int*)  d_in[2];
    const float* weight     = (const float*)d_in[3];
    float*       out        = (float*)d_out;

    const int n_nodes = in_sizes[2];          // 16384
    const int n_tiles = (n_nodes + 15) / 16;  // one block per 16-node tile

    element_update_wmma<<<n_tiles, 256, 0, stream>>>(h_prev, m_curr, atom_types,
                                                     weight, out);
}